// _ProposalTargetLayer_45921790329297
// MI455X (gfx1250) — compile-verified
//
#include <hip/hip_runtime.h>

#define NRR   6000
#define KK    20
#define FF    32
#define BEFF  256
#define NTT   6020      // NR + K
#define NSORT 8192
#define RPI   128
#define FGPI  32
#define NW    ((NTT + 31) / 32)
#define BD    1024

#define OFF_LAB 163840   // 256*128*5
#define OFF_BT  196608
#define OFF_IN  327680
#define OFF_OUT 458752

typedef unsigned int u32;
typedef unsigned long long u64;
typedef __attribute__((ext_vector_type(4))) u32 v4u;
typedef __attribute__((ext_vector_type(8))) int v8i;
typedef __attribute__((ext_vector_type(4))) int v4i;

#if __has_builtin(__builtin_amdgcn_tensor_load_to_lds)
#define HAVE_TDM 1
#else
#define HAVE_TDM 0
#endif

// --- Tensor Data Mover: 1-row 2D tile of `ndw` dwords, global -> LDS ----------
// D# layout per CDNA5 ISA ch.8 (group0: count/lds_addr/global_addr/type,
// group1: data_size=4B, tensor_dim0=ndw, tensor_dim1=1, tile_dim0=ndw,
// tile_dim1=1, stride0=ndw). Groups 2/3 (and the trailing int32x8 bank of the
// 6-arg clang-23 builtin) are zero: tensor is <=2D.
__device__ __forceinline__ void tdm_load_row(const float* src, float* lds_dst, int ndw) {
#if HAVE_TDM
  u64 ga = (u64)(uintptr_t)src;
  u32 la = (u32)(uintptr_t)lds_dst;   // low 32 bits of flat LDS addr == LDS byte offset
  v4u g0;
  g0[0] = 1u;                                          // count=1, gather off
  g0[1] = la;                                          // lds_addr
  g0[2] = (u32)(ga & 0xFFFFFFFFull);                   // global_addr[31:0]
  g0[3] = (u32)((ga >> 32) & 0x1FFFFFFull) | (2u << 30); // global_addr[56:32] | type=2
  u32 d0 = (u32)ndw;
  v8i g1;
  g1[0] = (int)(2u << 16);                             // data_size = 4 bytes
  g1[1] = (int)((d0 & 0xFFFFu) << 16);                 // tensor_dim0[15:0]
  g1[2] = (int)(((d0 >> 16) & 0xFFFFu) | (1u << 16));  // tensor_dim0[31:16] | tensor_dim1=1
  g1[3] = (int)((d0 & 0xFFFFu) << 16);                 // tile_dim0 = ndw
  g1[4] = 1;                                           // tile_dim1=1, tile_dim2=0
  g1[5] = (int)d0;                                     // tensor_dim0_stride[31:0]
  g1[6] = 0;
  g1[7] = 0;
  v4i gz4 = {0, 0, 0, 0};
  v8i gz8 = {0, 0, 0, 0, 0, 0, 0, 0};
  __builtin_amdgcn_tensor_load_to_lds(g0, g1, gz4, gz4, gz8, 0);
#else
  for (int t = (int)(threadIdx.x & 31); t < ndw; t += 32) lds_dst[t] = src[t];
#endif
}

__device__ __forceinline__ void tdm_wait() {
#if HAVE_TDM
  __builtin_amdgcn_s_wait_tensorcnt(0);
#endif
}

// --- roi coordinate fetch honoring the faithful permute/view scrambling ------
__device__ __forceinline__ void fetch_roi(const float* __restrict__ all_rois,
                                          const float* __restrict__ sgt,
                                          int img, int t,
                                          float& x1, float& y1, float& x2, float& y2) {
  if (t < NRR) {
    int g = img * NRR + t;
    int p = g / (NRR * FF);
    int q = (g % (NRR * FF)) / FF;
    const float* rp = all_rois + ((size_t)p * NRR + q) * 5;
    x1 = rp[1]; y1 = rp[2]; x2 = rp[3]; y2 = rp[4];
  } else {
    const float* gp = sgt + (size_t)(t - NRR) * 5;
    x1 = gp[0]; y1 = gp[1]; x2 = gp[2]; y2 = gp[3];
  }
}

__device__ __forceinline__ float iou_best(const float* __restrict__ sgt,
                                          const float* __restrict__ sga,
                                          const int* __restrict__ sgz,
                                          float bx1, float by1, float bx2, float by2,
                                          int* argp) {
  float bw = bx2 - bx1 + 1.f, bh = by2 - by1 + 1.f;
  float ba = bw * bh;
  bool bz = (bw == 1.f) && (bh == 1.f);
  float best = -INFINITY; int arg = 0;
  for (int k = 0; k < KK; ++k) {
    const float* g = sgt + k * 5;
    float w = fmaxf(fminf(bx2, g[2]) - fmaxf(bx1, g[0]) + 1.f, 0.f);
    float h = fmaxf(fminf(by2, g[3]) - fmaxf(by1, g[1]) + 1.f, 0.f);
    float inter = w * h;
    float ov = inter / (ba + sga[k] - inter);
    if (sgz[k]) ov = 0.f;
    if (bz)     ov = -1.f;
    if (ov > best) { best = ov; arg = k; }  // first-max semantics (JAX argmax)
  }
  *argp = arg;
  return best;
}

// --- stable ascending bitonic sort of (key, idx) pairs in LDS -----------------
__device__ __forceinline__ void bitonic_sort(float* key, unsigned short* idx, int tid) {
  for (int ksz = 2; ksz <= NSORT; ksz <<= 1) {
    for (int j = ksz >> 1; j > 0; j >>= 1) {
      for (int p = tid; p < NSORT / 2; p += BD) {
        int ii = ((p & ~(j - 1)) << 1) | (p & (j - 1));
        int jj = ii | j;
        bool up = ((ii & ksz) == 0);
        float ka = key[ii], kb = key[jj];
        unsigned short ia = idx[ii], ib = idx[jj];
        bool agtb = (ka > kb) || (ka == kb && ia > ib);  // idx tie-break == stable sort
        if (agtb == up) { key[ii] = kb; key[jj] = ka; idx[ii] = ib; idx[jj] = ia; }
      }
      __syncthreads();
    }
  }
}

__global__ __launch_bounds__(BD) void proposal_target_kernel(
    const float* __restrict__ all_rois,
    const float* __restrict__ gt_boxes,
    const float* __restrict__ pri_fg,
    const float* __restrict__ pri_bg,
    const float* __restrict__ u_fg,
    const float* __restrict__ u_bg,
    float* __restrict__ out) {
  const int img = blockIdx.x;
  const int tid = threadIdx.x;
  const int bi = img / FF, fi = img % FF;

  __shared__ float          s_key[NSORT];      // 32 KB: max_ov / pri staging / sort keys
  __shared__ unsigned short s_idx[NSORT];      // 16 KB
  __shared__ float          s_gt[KK * 5];
  __shared__ float          s_ga[KK];
  __shared__ int            s_gz[KK];
  __shared__ u32            s_fgb[NW];
  __shared__ u32            s_bgb[NW];
  __shared__ int            s_cnt[2];
  __shared__ int            s_pickfg[RPI];

  if (tid == 0) { s_cnt[0] = 0; s_cnt[1] = 0; }
  if (tid < KK * 5) {
    int k = tid / 5, c = tid % 5;
    s_gt[tid] = gt_boxes[((size_t)(bi * KK + k) * FF + fi) * 5 + c];
  }
  __syncthreads();
  if (tid < KK) {
    float gw = s_gt[tid * 5 + 2] - s_gt[tid * 5 + 0] + 1.f;
    float gh = s_gt[tid * 5 + 3] - s_gt[tid * 5 + 1] + 1.f;
    s_ga[tid] = gw * gh;
    s_gz[tid] = (gw == 1.f && gh == 1.f) ? 1 : 0;
  }
  __syncthreads();

  // Phase A: max IoU per roi -> s_key
  for (int t = tid; t < NTT; t += BD) {
    float x1, y1, x2, y2;
    fetch_roi(all_rois, s_gt, img, t, x1, y1, x2, y2);
    int arg;
    s_key[t] = iou_best(s_gt, s_ga, s_gz, x1, y1, x2, y2, &arg);
  }
  __syncthreads();

  // Phase B: fg/bg bitmasks + counts
  int cf = 0, cb = 0;
  for (int w = tid; w < NW; w += BD) {
    u32 fw = 0, bw = 0;
    int base = w * 32;
    for (int j = 0; j < 32; ++j) {
      int t = base + j;
      if (t < NTT) {
        float mo = s_key[t];
        if (mo >= 0.5f)       fw |= (1u << j);
        else if (mo >= 0.0f)  bw |= (1u << j);
      }
    }
    s_fgb[w] = fw; s_bgb[w] = bw;
    cf += __popc(fw); cb += __popc(bw);
  }
  atomicAdd(&s_cnt[0], cf);
  atomicAdd(&s_cnt[1], cb);
  __syncthreads();
  const int nfg = s_cnt[0], nbg = s_cnt[1];

  // Phase C: TDM-stage pri_fg row, build fg keys, sort
  if (tid < 32) { tdm_load_row(pri_fg + (size_t)img * NTT, s_key, NTT); tdm_wait(); }
  __syncthreads();
  for (int t = tid; t < NSORT; t += BD) {
    float key = 3.0f;
    if (t < NTT) key = ((s_fgb[t >> 5] >> (t & 31)) & 1u) ? s_key[t] : 2.0f;
    s_key[t] = key;
    s_idx[t] = (unsigned short)t;
  }
  __syncthreads();
  bitonic_sort(s_key, s_idx, tid);

  if (tid < RPI) {
    int fo = s_idx[tid];                                   // fg_order[slot]
    float ufv = u_fg[(size_t)img * RPI + tid];
    int rr = (int)floorf(ufv * (float)(nfg > 0 ? nfg : 1));
    rr = rr < 0 ? 0 : (rr >= NSORT ? NSORT - 1 : rr);
    int rep = s_idx[rr];                                   // fg_order[floor(u*nfg)]
    s_pickfg[tid] = (nbg > 0) ? fo : rep;
  }
  __syncthreads();

  // Phase D: TDM-stage pri_bg row, build bg keys, sort
  if (tid < 32) { tdm_load_row(pri_bg + (size_t)img * NTT, s_key, NTT); tdm_wait(); }
  __syncthreads();
  for (int t = tid; t < NSORT; t += BD) {
    float key = 3.0f;
    if (t < NTT) key = ((s_bgb[t >> 5] >> (t & 31)) & 1u) ? s_key[t] : 2.0f;
    s_key[t] = key;
    s_idx[t] = (unsigned short)t;
  }
  __syncthreads();
  bitonic_sort(s_key, s_idx, tid);

  // Phase E: resolve keep, gather, bbox-transform, write all five outputs
  if (tid < RPI) {
    int fg_take = (nfg > 0) ? ((nbg > 0) ? (FGPI < nfg ? FGPI : nfg) : RPI) : 0;
    float ubv = u_bg[(size_t)img * RPI + tid];
    int rb = (int)floorf(ubv * (float)(nbg > 0 ? nbg : 1));
    rb = rb < 0 ? 0 : (rb >= NSORT ? NSORT - 1 : rb);
    int bgp = s_idx[rb];
    int kp = (tid < fg_take) ? s_pickfg[tid] : bgp;

    float x1, y1, x2, y2;
    fetch_roi(all_rois, s_gt, img, kp, x1, y1, x2, y2);
    int ga;
    iou_best(s_gt, s_ga, s_gz, x1, y1, x2, y2, &ga);   // recompute gt_assign[kp]
    float gx1 = s_gt[ga * 5 + 0], gy1 = s_gt[ga * 5 + 1];
    float gx2 = s_gt[ga * 5 + 2], gy2 = s_gt[ga * 5 + 3];
    float glab = s_gt[ga * 5 + 4];
    float lab = (tid < fg_take) ? glab : 0.f;

    float ew = x2 - x1 + 1.f, eh = y2 - y1 + 1.f;
    float ecx = x1 + 0.5f * ew, ecy = y1 + 0.5f * eh;
    float gw = gx2 - gx1 + 1.f, gh = gy2 - gy1 + 1.f;
    float gcx = gx1 + 0.5f * gw, gcy = gy1 + 0.5f * gh;
    float t0 = ((gcx - ecx) / ew) / 0.1f;
    float t1 = ((gcy - ecy) / eh) / 0.1f;
    float t2 = logf(gw / ew) / 0.2f;
    float t3 = logf(gh / eh) / 0.2f;
    bool pos = lab > 0.f;
    float w1 = pos ? 1.f : 0.f;

    size_t row = (size_t)img * RPI + tid;
    float* ro = out + row * 5;
    ro[0] = (float)img; ro[1] = x1; ro[2] = y1; ro[3] = x2; ro[4] = y2;
    out[OFF_LAB + row] = lab;
    float* bt = out + OFF_BT + row * 4;
    bt[0] = pos ? t0 : 0.f; bt[1] = pos ? t1 : 0.f;
    bt[2] = pos ? t2 : 0.f; bt[3] = pos ? t3 : 0.f;
    float* iw = out + OFF_IN + row * 4;
    iw[0] = w1; iw[1] = w1; iw[2] = w1; iw[3] = w1;
    float* ow = out + OFF_OUT + row * 4;
    ow[0] = w1; ow[1] = w1; ow[2] = w1; ow[3] = w1;
  }
}

extern "C" void kernel_launch(void* const* d_in, const int* in_sizes, int n_in,
                              void* d_out, int out_size, void* d_ws, size_t ws_size,
                              hipStream_t stream) {
  (void)in_sizes; (void)n_in; (void)d_ws; (void)ws_size; (void)out_size;
  const float* all_rois = (const float*)d_in[0];
  const float* gt_boxes = (const float*)d_in[1];
  // d_in[2] = num_boxes (unused by the reference)
  const float* pri_fg = (const float*)d_in[3];
  const float* pri_bg = (const float*)d_in[4];
  const float* u_fg   = (const float*)d_in[5];
  const float* u_bg   = (const float*)d_in[6];
  float* out = (float*)d_out;

  proposal_target_kernel<<<dim3(BEFF), dim3(BD), 0, stream>>>(
      all_rois, gt_boxes, pri_fg, pri_bg, u_fg, u_bg, out);
}